// ConditionedCategorical_34368328303367
// MI455X (gfx1250) — compile-verified
//
#include <hip/hip_runtime.h>

// ConditionedCategorical M-step on MI455X (gfx1250, wave32).
//
// Roofline: 288 MB of reads / 23.3 TB/s = ~12.4 us -> memory-bound scatter-add.
// Path: coalesced B128 loads -> LDS-privatized histogram (bank-padded, ds_add_f32)
//       -> global f32 atomic merge -> WMMA(f32 16x16x4) Y-reduction + normalize.

#define KK   16
#define YY   32
#define CC   16
#define NSEG (KK * YY)      // 512 composite bins
#define PAD  17             // padded LDS row stride (floats) -> spreads 64 banks
#define EPSV 1e-8f

typedef __attribute__((ext_vector_type(2))) float v2f;
typedef __attribute__((ext_vector_type(8))) float v8f;

// ---------------------------------------------------------------- kernel 0 ---
__global__ void zero_ws_kernel(float* __restrict__ ws) {
    int i = blockIdx.x * 256 + threadIdx.x;
    if (i < NSEG * CC) ws[i] = 0.0f;
}

// ---------------------------------------------------------------- kernel 1 ---
// Each thread consumes whole samples: 4x float4 loads (wave reads 2KB
// contiguous -> full HBM efficiency), 16x ds_add_f32 into the padded LDS
// histogram. comp*17 + c walks all 64 LDS banks for random comp, so the
// expected conflict degree over 32 lanes is ~2-3x instead of 8x at stride 16.
__global__ __launch_bounds__(256) void hist_kernel(
    const int*    __restrict__ xl,
    const int*    __restrict__ yl,
    const float4* __restrict__ post,   // N rows, 4 float4 per row
    float*        __restrict__ ws,     // [512*16] global accumulator
    int n)
{
    __shared__ float hist[NSEG * PAD];
    for (int i = threadIdx.x; i < NSEG * PAD; i += 256) hist[i] = 0.0f;
    __syncthreads();

    const int stride = gridDim.x * 256;
    for (int s = blockIdx.x * 256 + threadIdx.x; s < n; s += stride) {
        const int comp = xl[s] * YY + yl[s];          // [0, 512)
        const float4 p0 = post[s * 4 + 0];
        const float4 p1 = post[s * 4 + 1];
        const float4 p2 = post[s * 4 + 2];
        const float4 p3 = post[s * 4 + 3];
        float* b = &hist[comp * PAD];
        atomicAdd(b +  0, p0.x); atomicAdd(b +  1, p0.y);
        atomicAdd(b +  2, p0.z); atomicAdd(b +  3, p0.w);
        atomicAdd(b +  4, p1.x); atomicAdd(b +  5, p1.y);
        atomicAdd(b +  6, p1.z); atomicAdd(b +  7, p1.w);
        atomicAdd(b +  8, p2.x); atomicAdd(b +  9, p2.y);
        atomicAdd(b + 10, p2.z); atomicAdd(b + 11, p2.w);
        atomicAdd(b + 12, p3.x); atomicAdd(b + 13, p3.y);
        atomicAdd(b + 14, p3.z); atomicAdd(b + 15, p3.w);
    }
    __syncthreads();

    // Merge private histogram into the global accumulator (8192 atomics / WG).
    for (int i = threadIdx.x; i < NSEG * CC; i += 256) {
        const float v = hist[(i >> 4) * PAD + (i & 15)];
        unsafeAtomicAdd(&ws[i], v);
    }
}

// ---------------------------------------------------------------- kernel 2 ---
// denom[k][c] = sum_y num[k][y][c], computed per wave (wave w <-> k=w) as
// 8 chained V_WMMA_F32_16X16X4_F32 with A == ones(16x4): D += ones * B gives
// exact f32 column sums (every D row identical). Then out = (num+eps)/denom.
__global__ __launch_bounds__(512) void norm_kernel(
    const float* __restrict__ num,     // [512*16]
    float*       __restrict__ out)     // [16*32*16]
{
    __shared__ float denom_s[KK * CC];

    const int tid  = threadIdx.x;
    const int w    = tid >> 5;         // wave id == k  (16 waves)
    const int lane = tid & 31;
    const int col  = lane & 15;        // B/D column N = lane%16
    const int krow = (lane >> 4) << 1; // B rows: lanes 0-15 -> K={0,1}, 16-31 -> K={2,3}

    v2f a; a.x = 1.0f; a.y = 1.0f;     // ones 16x4 A fragment (all lanes, both VGPRs)
    v8f acc = {};

    const float* base = num + w * (YY * CC);
#pragma unroll
    for (int yb = 0; yb < YY; yb += 4) {
        v2f b;
        b.x = base[(yb + krow + 0) * CC + col];
        b.y = base[(yb + krow + 1) * CC + col];
        acc = __builtin_amdgcn_wmma_f32_16x16x4_f32(
                  /*neg_a=*/false, a, /*neg_b=*/false, b,
                  /*c_mod=*/(short)0, acc, /*reuse_a=*/false, /*reuse_b=*/false);
    }

    // D row 0 lives in acc[0]: lanes 0-15 carry column sums for c = lane.
    if (lane < 16) denom_s[w * CC + col] = acc[0] + (float)YY * EPSV;
    __syncthreads();

    for (int i = tid; i < KK * YY * CC; i += 512) {
        const int k = i >> 9;          // /(YY*CC)
        const int c = i & 15;
        out[i] = (num[i] + EPSV) / denom_s[k * CC + c];
    }
}

// ------------------------------------------------------------------ launch ---
extern "C" void kernel_launch(void* const* d_in, const int* in_sizes, int n_in,
                              void* d_out, int out_size, void* d_ws, size_t ws_size,
                              hipStream_t stream) {
    const int*   xl   = (const int*)  d_in[0];
    const int*   yl   = (const int*)  d_in[1];
    const float* post = (const float*)d_in[2];
    float*       out  = (float*)d_out;
    float*       ws   = (float*)d_ws;          // needs 512*16*4 = 32 KB
    const int    n    = in_sizes[0];

    zero_ws_kernel<<<(NSEG * CC + 255) / 256, 256, 0, stream>>>(ws);
    hist_kernel<<<512, 256, 0, stream>>>(xl, yl, (const float4*)post, ws, n);
    norm_kernel<<<1, 512, 0, stream>>>(ws, out);
}